// SFMSeq_78580721647915
// MI455X (gfx1250) — compile-verified
//
#include <hip/hip_runtime.h>
#include <math.h>

// Problem dims (fixed by the reference)
#define NSTEPS 4096   // B*T, fully sequential scan
#define INPUTD 128
#define HID    256
#define NFREQ  64
#define NOUT   128
#define NWG    16          // cooperative workgroups in the scan (HID/16)
#define RPW    16          // hidden rows per workgroup
#define SCAN_THREADS 256
#define U4S    260         // LDS row stride: 16B-aligned rows + bank rotation
#define WAS    65

typedef __attribute__((ext_vector_type(2))) float v2f;
typedef __attribute__((ext_vector_type(8))) float v8f;

__device__ __forceinline__ float sigm_(float x) { return 1.0f / (1.0f + expf(-x)); }

// Stage 16 bytes global -> LDS via the gfx1250 async DMA path (ASYNCcnt).
// vdst = LDS byte address (low 32 bits of generic LDS pointer), vaddr = 64-bit
// global address, saddr = off (GV mode).  Inline asm avoids the clang builtin
// whose signature differs across toolchains.
__device__ __forceinline__ void stage16(const float* __restrict__ gsrc, float* ldst)
{
  unsigned lds_addr = (unsigned)(size_t)ldst;
  unsigned long long gaddr = (unsigned long long)(size_t)gsrc;
  asm volatile("global_load_async_to_lds_b128 %0, %1, off"
               :: "v"(lds_addr), "v"(gaddr)
               : "memory");
}

__device__ __forceinline__ void stage_wait()
{
#if __has_builtin(__builtin_amdgcn_s_wait_asynccnt)
  __builtin_amdgcn_s_wait_asynccnt(0);
#else
  asm volatile("s_wait_asynccnt 0" ::: "memory");
#endif
}

// ---------------------------------------------------------------------------
// WMMA GEMM with bias: C[M x N] = A[M x Kd] * W[N x Kd]^T + bias[N]
// fp32 WMMA 16x16x4; one wave computes one 16x16 tile; 4 waves/block.
// A tile layout (ISA 7.12.2, 32-bit A 16x4): lane<16 -> M=lane, K={k0,k0+1};
// lane>=16 -> M=lane-16, K={k0+2,k0+3}.  B mirrors it (row striped in lanes).
// C/D: VGPR r: lanes0-15 M=r, lanes16-31 M=r+8, N=lane&15.
// ---------------------------------------------------------------------------
__global__ __launch_bounds__(128) void wmma_gemm_bias(
    const float* __restrict__ A, const float* __restrict__ W,
    const float* __restrict__ bias, float* __restrict__ C,
    int M, int Kd, int N)
{
  const int lane = threadIdx.x & 31;
  const int wave = threadIdx.x >> 5;
  const int lm   = lane & 15;
  const int half = lane >> 4;
  const int t0 = blockIdx.x * 16;
  const int j0 = blockIdx.y * 64 + wave * 16;
  if (t0 >= M || j0 >= N) return;            // wave-uniform: EXEC stays all-1s

  const float* arow = A + (size_t)(t0 + lm) * Kd + half * 2;
  const float* wrow = W + (size_t)(j0 + lm) * Kd + half * 2;

  v8f acc = {};
  for (int k0 = 0; k0 < Kd; k0 += 4) {
    v2f a, b;
    a.x = arow[k0]; a.y = arow[k0 + 1];
    b.x = wrow[k0]; b.y = wrow[k0 + 1];
    acc = __builtin_amdgcn_wmma_f32_16x16x4_f32(false, a, false, b,
                                                (short)0, acc, false, false);
  }
  const float bj = bias[j0 + lm];
#pragma unroll
  for (int r = 0; r < 8; ++r) {
    int row = t0 + r + half * 8;
    C[(size_t)row * N + (j0 + lm)] = acc[r] + bj;
  }
}

// ---------------------------------------------------------------------------
// Init: zero h0 row and the global barrier state (ws is poisoned 0xAA).
// ---------------------------------------------------------------------------
__global__ void scan_init(float* __restrict__ hall, int* __restrict__ bar)
{
  int i = threadIdx.x;
  if (i < HID) hall[i] = 0.0f;
  if (i < 2) bar[i] = 0;
}

// ---------------------------------------------------------------------------
// Persistent cooperative scan over NSTEPS.
// WG w owns hidden rows [w*16, w*16+16).  Recurrent weights live in LDS,
// re/im state in registers (4 freq bins per thread).  Cross-WGP h exchange
// via agent-scope atomics + epoch barrier.
// ---------------------------------------------------------------------------
__global__ __launch_bounds__(SCAN_THREADS) void sfm_scan(
    const float* __restrict__ Uste, const float* __restrict__ Ufre,
    const float* __restrict__ Ug,   const float* __restrict__ Ui,
    const float* __restrict__ Uo,   const float* __restrict__ Wa,
    const float* __restrict__ ba,
    const float* __restrict__ Pste, const float* __restrict__ Pfre,
    const float* __restrict__ Pg,   const float* __restrict__ Pi,
    const float* __restrict__ Po,
    float* __restrict__ hall, int* __restrict__ bar)
{
  extern __shared__ float smem[];
  float* u4     = smem;                 // 64 rows (ste,g,i,o x 16) x 260
  float* ufre   = u4   + 64 * U4S;      // 64 rows x 260 (full Ufre, replicated)
  float* wa_s   = ufre + 64 * U4S;      // 16 x 65
  float* h_s    = wa_s + RPW * WAS;     // 256
  float* part   = h_s  + HID;           // 256
  float* ffre_s = part + SCAN_THREADS;  // 64
  float* fste_s = ffre_s + NFREQ;       // 16
  float* gg_s   = fste_s + RPW;         // 16
  float* ii_s   = gg_s + RPW;           // 16
  float* oo_s   = ii_s + RPW;           // 16
  float* gi_s   = oo_s + RPW;           // 16
  float* ba_s   = gi_s + RPW;           // 16
  float* ctab   = ba_s + RPW;           // 64
  float* stab   = ctab + NFREQ;         // 64

  const int tid   = threadIdx.x;
  const int wg    = blockIdx.x;
  const int hbase = wg * RPW;

  // ---- one-time LDS staging (async global->LDS DMA, float4 granules) ----
  const float* Usrc[4] = {Uste, Ug, Ui, Uo};
#pragma unroll
  for (int g = 0; g < 4; ++g) {
    const float* src = Usrc[g];
    for (int e = tid; e < RPW * 64; e += SCAN_THREADS) {   // 16 rows x 64 float4
      int r = e >> 6, c4 = (e & 63) << 2;
      stage16(&src[(size_t)(hbase + r) * HID + c4], &u4[(g * 16 + r) * U4S + c4]);
    }
  }
  for (int e = tid; e < NFREQ * 64; e += SCAN_THREADS) {   // 64 rows x 64 float4
    int r = e >> 6, c4 = (e & 63) << 2;
    stage16(&Ufre[(size_t)r * HID + c4], &ufre[r * U4S + c4]);
  }
  for (int e = tid; e < RPW * NFREQ; e += SCAN_THREADS) {  // small, plain loads
    int r = e >> 6, c = e & 63;
    wa_s[r * WAS + c] = Wa[(size_t)(hbase + r) * NFREQ + c];
  }
  if (tid < RPW) ba_s[tid] = ba[hbase + tid];
  if (tid < NFREQ) {
    // cos(omega_k * t) = cos(2*pi*(k*t mod 64)/64) -> exact 64-entry table
    float ang = 6.2831853071795864769f * (float)tid / 64.0f;
    ctab[tid] = cosf(ang);
    stab[tid] = sinf(ang);
  }
  stage_wait();
  __syncthreads();

  // per-thread frequency state: row hh, freq bins kq..kq+3
  const int hh = tid >> 4;
  const int kq = (tid & 15) << 2;
  float re[4] = {0.f, 0.f, 0.f, 0.f};
  float im[4] = {0.f, 0.f, 0.f, 0.f};

  // dot-product assignment: output j = tid/2 (0..63: f_fre, 64..127: gates),
  // each half-thread sums 128 of the 256 h elements.  Rows are 16B aligned
  // (stride 260 floats), so the dot runs on ds_load_b128.
  const int j    = tid >> 1;
  const int half = tid & 1;
  const float4* dotrow = (const float4*)((j < 64)
                            ? &ufre[j * U4S + half * 128]
                            : &u4[(j - 64) * U4S + half * 128]);
  const float4* hvec = (const float4*)&h_s[half * 128];

  for (int t = 0; t < NSTEPS; ++t) {
    // h_{t-1} (agent scope: produced by other WGPs last step)
    if (tid < HID)
      h_s[tid] = __hip_atomic_load(&hall[(size_t)t * HID + tid],
                                   __ATOMIC_RELAXED, __HIP_MEMORY_SCOPE_AGENT);
    // prefetch next step's gate preactivations while this step computes
    if (tid == 0 && t + 1 < NSTEPS) {
      size_t tn = (size_t)(t + 1);
      __builtin_prefetch(&Pfre[tn * NFREQ], 0, 1);
      __builtin_prefetch(&Pfre[tn * NFREQ + 32], 0, 1);
      __builtin_prefetch(&Pste[tn * HID + hbase], 0, 1);
      __builtin_prefetch(&Pg[tn * HID + hbase], 0, 1);
      __builtin_prefetch(&Pi[tn * HID + hbase], 0, 1);
      __builtin_prefetch(&Po[tn * HID + hbase], 0, 1);
    }
    __syncthreads();

    // U @ h partial dots (float4 LDS loads)
    float p0 = 0.f, p1 = 0.f, p2 = 0.f, p3 = 0.f;
#pragma unroll 8
    for (int kk = 0; kk < 32; ++kk) {
      float4 a = dotrow[kk];
      float4 b = hvec[kk];
      p0 = fmaf(a.x, b.x, p0);
      p1 = fmaf(a.y, b.y, p1);
      p2 = fmaf(a.z, b.z, p2);
      p3 = fmaf(a.w, b.w, p3);
    }
    part[tid] = (p0 + p1) + (p2 + p3);
    __syncthreads();

    if (tid < 128) {
      float s = part[2 * tid] + part[2 * tid + 1];
      if (tid < 64) {
        ffre_s[tid] = sigm_(s + Pfre[(size_t)t * NFREQ + tid]);  // bias folded in P
      } else {
        int idx = tid - 64, g = idx >> 4, r = idx & 15;
        const float* Pp = (g == 0) ? Pste : (g == 1) ? Pg : (g == 2) ? Pi : Po;
        float pre = s + Pp[(size_t)t * HID + hbase + r];
        if (g == 2)      ii_s[r]   = tanhf(pre);
        else if (g == 0) fste_s[r] = sigm_(pre);
        else if (g == 1) gg_s[r]   = sigm_(pre);
        else             oo_s[r]   = sigm_(pre);
      }
    }
    __syncthreads();
    if (tid < RPW) gi_s[tid] = gg_s[tid] * ii_s[tid];
    __syncthreads();

    // frequency-state update + amplitude combine
    float acc = 0.f;
    const float fs = fste_s[hh];
    const float gv = gi_s[hh];
    const unsigned tt = (unsigned)(t + 1);
#pragma unroll
    for (int u = 0; u < 4; ++u) {
      int k = kq + u;
      unsigned m = ((unsigned)k * tt) & 63u;
      float Ft = fs * ffre_s[k];
      re[u] = fmaf(Ft, re[u], gv * ctab[m]);
      im[u] = fmaf(Ft, im[u], gv * stab[m]);
      float amp = sqrtf(fmaf(re[u], re[u], fmaf(im[u], im[u], 1e-8f)));
      acc = fmaf(amp, wa_s[hh * WAS + k], acc);
    }
    part[tid] = acc;
    __syncthreads();
    if (tid < RPW) {
      float s = 0.f;
#pragma unroll
      for (int q = 0; q < 16; ++q) s += part[tid * 16 + q];
      float c = tanhf(s + ba_s[tid]);
      float hnew = oo_s[tid] * c;
      __hip_atomic_store(&hall[(size_t)(t + 1) * HID + hbase + tid], hnew,
                         __ATOMIC_RELAXED, __HIP_MEMORY_SCOPE_AGENT);
    }

    // device-scope epoch barrier (monotonic sense, no ABA)
    __syncthreads();
    if (tid == 0) {
      __threadfence();
      int arrived = __hip_atomic_fetch_add(&bar[0], 1, __ATOMIC_ACQ_REL,
                                           __HIP_MEMORY_SCOPE_AGENT);
      if (arrived == NWG - 1) {
        __hip_atomic_store(&bar[0], 0, __ATOMIC_RELAXED, __HIP_MEMORY_SCOPE_AGENT);
        __hip_atomic_store(&bar[1], t + 1, __ATOMIC_RELEASE, __HIP_MEMORY_SCOPE_AGENT);
      } else {
        while (__hip_atomic_load(&bar[1], __ATOMIC_ACQUIRE,
                                 __HIP_MEMORY_SCOPE_AGENT) < t + 1)
          __builtin_amdgcn_s_sleep(1);
      }
    }
    __syncthreads();
  }
}

// ---------------------------------------------------------------------------
extern "C" void kernel_launch(void* const* d_in, const int* in_sizes, int n_in,
                              void* d_out, int out_size, void* d_ws, size_t ws_size,
                              hipStream_t stream)
{
  (void)in_sizes; (void)n_in; (void)out_size; (void)ws_size;

  const float* X    = (const float*)d_in[0];
  const float* Wste = (const float*)d_in[1];
  const float* Uste = (const float*)d_in[2];
  const float* bste = (const float*)d_in[3];
  const float* Wfre = (const float*)d_in[4];
  const float* Ufre = (const float*)d_in[5];
  const float* bfre = (const float*)d_in[6];
  const float* Wg   = (const float*)d_in[7];
  const float* Ug   = (const float*)d_in[8];
  const float* bg   = (const float*)d_in[9];
  const float* Wi   = (const float*)d_in[10];
  const float* Ui   = (const float*)d_in[11];
  const float* bi   = (const float*)d_in[12];
  const float* Wo   = (const float*)d_in[13];
  const float* Uo   = (const float*)d_in[14];
  const float* bo   = (const float*)d_in[15];
  const float* Wa   = (const float*)d_in[16];
  const float* ba   = (const float*)d_in[17];
  const float* Wout = (const float*)d_in[18];
  const float* bout = (const float*)d_in[19];
  const float* Wfc  = (const float*)d_in[20];
  const float* bfc  = (const float*)d_in[21];
  float* out = (float*)d_out;

  // workspace layout (floats)
  float* ws   = (float*)d_ws;
  int*   bar  = (int*)ws;                                 // 64B reserved
  float* hall = ws + 16;                                  // (4097)*256: h_{-1..4095}
  float* Ps   = hall + (size_t)(NSTEPS + 1) * HID;        // 4096*256
  float* Pgg  = Ps   + (size_t)NSTEPS * HID;              // 4096*256
  float* Pii  = Pgg  + (size_t)NSTEPS * HID;              // 4096*256
  float* Poo  = Pii  + (size_t)NSTEPS * HID;              // 4096*256
  float* Pff  = Poo  + (size_t)NSTEPS * HID;              // 4096*64
  float* OutB = Pff  + (size_t)NSTEPS * NFREQ;            // 4096*128

  const dim3 blk(128);
  const dim3 gH (NSTEPS / 16, HID   / 64);
  const dim3 gF (NSTEPS / 16, NFREQ / 64);
  const dim3 gO (NSTEPS / 16, NOUT  / 64);

  // Phase 1: input-side projections (all timesteps in parallel, fp32 WMMA)
  wmma_gemm_bias<<<gH, blk, 0, stream>>>(X, Wste, bste, Ps,  NSTEPS, INPUTD, HID);
  wmma_gemm_bias<<<gH, blk, 0, stream>>>(X, Wg,   bg,   Pgg, NSTEPS, INPUTD, HID);
  wmma_gemm_bias<<<gH, blk, 0, stream>>>(X, Wi,   bi,   Pii, NSTEPS, INPUTD, HID);
  wmma_gemm_bias<<<gH, blk, 0, stream>>>(X, Wo,   bo,   Poo, NSTEPS, INPUTD, HID);
  wmma_gemm_bias<<<gF, blk, 0, stream>>>(X, Wfre, bfre, Pff, NSTEPS, INPUTD, NFREQ);

  // Phase 2: sequential scan (persistent cooperative kernel, 16 WGPs)
  scan_init<<<1, 256, 0, stream>>>(hall, bar);
  const size_t smemBytes =
      (2 * 64 * U4S + RPW * WAS + HID + SCAN_THREADS + NFREQ + 6 * RPW + 2 * NFREQ)
      * sizeof(float);
  sfm_scan<<<NWG, SCAN_THREADS, smemBytes, stream>>>(
      Uste, Ufre, Ug, Ui, Uo, Wa, ba, Ps, Pff, Pgg, Pii, Poo, hall, bar);

  // Phase 3: output projection + trailing FC (WMMA again)
  wmma_gemm_bias<<<gO, blk, 0, stream>>>(hall + HID, Wout, bout, OutB, NSTEPS, HID, NOUT);
  wmma_gemm_bias<<<gO, blk, 0, stream>>>(OutB, Wfc, bfc, out, NSTEPS, NOUT, NOUT);
}